// ChannelAttention1D_82197084111350
// MI455X (gfx1250) — compile-verified
//
#include <hip/hip_runtime.h>

typedef float v2f __attribute__((ext_vector_type(2)));
typedef float v4f __attribute__((ext_vector_type(4)));
typedef float v8f __attribute__((ext_vector_type(8)));

// Problem constants (from reference)
constexpr int BATCH = 4096;
constexpr int NF    = 64;    // frames / channels
constexpr int ND    = 512;   // feature dim
constexpr int NR    = 16;    // reduced dim (F/RED)
constexpr int NK    = 8;     // top-k
constexpr int BPB   = 16;    // batches per block (one WMMA M-tile)

__global__ __launch_bounds__(256)
void chanattn1d_kernel(const float* __restrict__ x,
                       const float* __restrict__ w1,
                       const float* __restrict__ b1,
                       const float* __restrict__ w2,
                       const float* __restrict__ b2,
                       float* __restrict__ out)
{
    __shared__ float pooled_lds[BPB * NF];  // 1024 f32
    __shared__ float h_lds[BPB * NR];       // 256 f32
    __shared__ float w_lds[BPB * NF];       // 1024 f32 (channel weights, consumed by topk)
    __shared__ int   idx_lds[BPB * NK];     // 128 int

    const int tid  = threadIdx.x;
    const int wave = tid >> 5;
    const int lane = tid & 31;
    const int hf   = (lane >> 4) & 1;   // half-wave id
    const int sub  = lane & 15;         // lane within half
    const int b0   = blockIdx.x * BPB;

    // ---------------- Stage 1: mean over D per (b,f) row ----------------
    // 1024 rows of 512 f32; one row per half-wave, 16B coalesced loads,
    // 4-step shfl_xor reduction within the 16-lane half.
    {
        const float* xblk = x + (size_t)b0 * NF * ND;
        for (int p = wave; p < (BPB * NF) / 2; p += 8) {
            int rb = p * 2 + hf;                 // row-in-block: b_local*64 + f
            const v4f* src = (const v4f*)(xblk + (size_t)rb * ND);
            float s = 0.f;
            #pragma unroll
            for (int i = 0; i < 8; ++i) {
                v4f v = src[sub + i * 16];
                s += v.x + v.y + v.z + v.w;
            }
            s += __shfl_xor(s, 1);
            s += __shfl_xor(s, 2);
            s += __shfl_xor(s, 4);
            s += __shfl_xor(s, 8);
            if (sub == 0) pooled_lds[rb] = s * (1.0f / (float)ND);
        }
    }
    __syncthreads();

    // ---------------- Stage 2: MLP via V_WMMA_F32_16X16X4_F32 (wave 0) ----------------
    if (tid < 32) {
        // Layer 1: h[16x16] = relu(pooled[16x64] x w1^T[64x16] + b1)
        // A 16x4 f32 frag: lanes 0-15 hold (M=sub, K=kb..kb+1), lanes 16-31 K=kb+2..kb+3
        // B 4x16 f32 frag: B[k][n] = w1[n][k]
        v8f acc = {};
        #pragma unroll
        for (int kk = 0; kk < 16; ++kk) {
            int kb = kk * 4 + hf * 2;
            v2f a, b;
            a.x = pooled_lds[sub * NF + kb];
            a.y = pooled_lds[sub * NF + kb + 1];
            b.x = w1[sub * NF + kb];
            b.y = w1[sub * NF + kb + 1];
            acc = __builtin_amdgcn_wmma_f32_16x16x4_f32(
                false, a, false, b, (short)0, acc, false, false);
        }
        // C/D layout: acc[v] = h[M = v + hf*8][N = sub]
        float bias1 = b1[sub];
        #pragma unroll
        for (int v = 0; v < 8; ++v) {
            float hv = acc[v] + bias1;
            hv = hv > 0.f ? hv : 0.f;
            h_lds[(v + hf * 8) * NR + sub] = hv;   // h row-major [16 batch][16 r]
        }
        // same-wave LDS RAW: wait for DS writes before re-reading as A-frag
        asm volatile("s_wait_dscnt 0" ::: "memory");

        // Layer 2: weights[16x64] = sigmoid(h[16x16] x w2^T[16x64] + b2)
        #pragma unroll
        for (int nt = 0; nt < 4; ++nt) {
            v8f acc2 = {};
            #pragma unroll
            for (int kk = 0; kk < 4; ++kk) {
                int kb = kk * 4 + hf * 2;
                v2f a, b;
                a.x = h_lds[sub * NR + kb];
                a.y = h_lds[sub * NR + kb + 1];
                int f = nt * 16 + sub;             // B[k=r][n=f] = w2[f][r]
                b.x = w2[f * NR + kb];
                b.y = w2[f * NR + kb + 1];
                acc2 = __builtin_amdgcn_wmma_f32_16x16x4_f32(
                    false, a, false, b, (short)0, acc2, false, false);
            }
            int f = nt * 16 + sub;
            float bias2 = b2[f];
            #pragma unroll
            for (int v = 0; v < 8; ++v) {
                float wv = acc2[v] + bias2;
                wv = 1.0f / (1.0f + __expf(-wv));
                w_lds[(v + hf * 8) * NF + f] = wv;
            }
        }
    }
    __syncthreads();

    // ---------------- Stage 3: top-8 of 64 per batch (16 threads) ----------------
    if (tid < BPB) {
        float* wr = &w_lds[tid * NF];
        #pragma unroll 1
        for (int k = 0; k < NK; ++k) {
            float best = -3.4e38f;
            int   bi   = 0;
            for (int j = 0; j < NF; ++j) {
                float v = wr[j];
                if (v > best) { best = v; bi = j; }   // first occurrence wins (jax tie rule)
            }
            wr[bi] = -3.4e38f;
            idx_lds[tid * NK + k] = bi;
        }
    }
    __syncthreads();

    // ---------------- Stage 4: gather selected frames ----------------
    // 128 output rows of 512 f32 per block; source is L2-hot (just streamed).
    // Non-temporal stores keep the 67MB output from evicting x in L2.
    for (int p = wave; p < (BPB * NK) / 2; p += 8) {
        int q  = p * 2 + hf;
        int bl = q >> 3;
        int kk = q & 7;
        int idx = idx_lds[bl * NK + kk];
        const v4f* src = (const v4f*)(x + (((size_t)(b0 + bl) * NF + idx) * ND));
        v4f*       dst = (v4f*)(out + (((size_t)(b0 + bl) * NK + kk) * ND));
        #pragma unroll
        for (int i = 0; i < 8; ++i) {
            v4f v = src[sub + i * 16];
            __builtin_nontemporal_store(v, &dst[sub + i * 16]);
        }
    }
}

extern "C" void kernel_launch(void* const* d_in, const int* in_sizes, int n_in,
                              void* d_out, int out_size, void* d_ws, size_t ws_size,
                              hipStream_t stream) {
    const float* x  = (const float*)d_in[0];
    const float* w1 = (const float*)d_in[1];
    const float* b1 = (const float*)d_in[2];
    const float* w2 = (const float*)d_in[3];
    const float* b2 = (const float*)d_in[4];
    float* out = (float*)d_out;

    dim3 grid(BATCH / BPB);   // 256 blocks, each owns 16 batches
    dim3 block(256);          // 8 waves
    chanattn1d_kernel<<<grid, block, 0, stream>>>(x, w1, b1, w2, b2, out);
}